// GCN_69097433858735
// MI455X (gfx1250) — compile-verified
//
#include <hip/hip_runtime.h>

typedef __attribute__((ext_vector_type(2))) float v2f;
typedef __attribute__((ext_vector_type(8))) float v8f;

#define TPB 256

static __device__ __forceinline__ void atomAddF32(float* p, float v) {
  // Should lower to global_atomic_add_f32 (no-return) on gfx1250.
  unsafeAtomicAdd(p, v);
}

// ---- degree / norm -------------------------------------------------------

__global__ void k_init_deg(float* __restrict__ deg, int N) {
  int n = blockIdx.x * blockDim.x + threadIdx.x;
  if (n < N) deg[n] = 1.0f;  // self-loop contribution
}

__global__ void k_edge_deg(const int* __restrict__ dst, float* __restrict__ deg, int E) {
  int e = blockIdx.x * blockDim.x + threadIdx.x;
  if (e < E) atomAddF32(&deg[dst[e]], 1.0f);
}

__global__ void k_dinv(float* __restrict__ deg, int N) {
  int n = blockIdx.x * blockDim.x + threadIdx.x;
  if (n < N) {
    float d = deg[n];
    deg[n] = (d > 0.0f) ? rsqrtf(d) : 0.0f;  // deg >= 1 always, but match ref
  }
}

// ---- layer 1 dense transform: h1 = x @ W1, agg1 init = h1 * dinv^2 -------
// One wave computes a 16(node)x16(hidden) tile with V_WMMA_F32_16X16X4_F32,
// looping K=128 in steps of 4.
//
// A (16x4 f32) per ISA: lanes 0-15 -> M=lane, K={k0,k0+1}; lanes 16-31 ->
// M=lane-16, K={k0+2,k0+3}. B (4x16) mirrored: VGPR0/1 hold rows k0+2*half,
// k0+2*half+1 striped across the 16 columns.

__global__ void k_gemm1_wmma(const float* __restrict__ x,
                             const float* __restrict__ W1,
                             const float* __restrict__ dinv,
                             float* __restrict__ h1,
                             float* __restrict__ agg1, int N) {
  const int lane = threadIdx.x & 31;
  const int tile = blockIdx.x * (blockDim.x >> 5) + (threadIdx.x >> 5);
  const int row0 = tile * 16;
  if (row0 >= N) return;  // wave-uniform: EXEC stays all-ones for WMMA

  const int half = lane >> 4;   // 0 or 1 -> +2 in K
  const int lrow = lane & 15;

  int arow = row0 + lrow;
  if (arow >= N) arow = N - 1;  // clamp loads; stores guarded below

  const float* xrow = x + (size_t)arow * 128 + half * 2;
  const float* wcol = W1 + (size_t)(half * 2) * 16 + lrow;

  v8f acc = {};
#pragma unroll
  for (int k0 = 0; k0 < 128; k0 += 4) {
    v2f a = *(const v2f*)(xrow + k0);       // b64 load, 8B aligned
    v2f b;
    b.x = wcol[(size_t)k0 * 16];
    b.y = wcol[(size_t)k0 * 16 + 16];
    acc = __builtin_amdgcn_wmma_f32_16x16x4_f32(
        /*neg_a=*/false, a, /*neg_b=*/false, b,
        /*c_mod=*/(short)0, acc, /*reuse_a=*/false, /*reuse_b=*/false);
  }

  // C/D layout: VGPR v -> M = v + 8*half, N = lane&15
  const int col = lrow;
#pragma unroll
  for (int v = 0; v < 8; ++v) {
    const int m = row0 + v + half * 8;
    if (m < N) {
      const float val = acc[v];
      h1[(size_t)m * 16 + col] = val;
      const float di = dinv[m];
      agg1[(size_t)m * 16 + col] = val * di * di;  // self-loop term
    }
  }
}

// ---- layer 1 edge aggregation: agg1[dst] += h1[src] * norm ---------------

__global__ void k_edge_agg16(const int* __restrict__ ei,
                             const float* __restrict__ dinv,
                             const float* __restrict__ h1,
                             float* __restrict__ agg1, int E) {
  int e = blockIdx.x * blockDim.x + threadIdx.x;
  if (e >= E) return;
  const int s = ei[e];
  const int d = ei[E + e];
  const float w = dinv[s] * dinv[d];
  const float4* hs = (const float4*)(h1 + (size_t)s * 16);
  float4 m0 = hs[0], m1 = hs[1], m2 = hs[2], m3 = hs[3];
  float* op = agg1 + (size_t)d * 16;
  atomAddF32(op + 0,  m0.x * w);
  atomAddF32(op + 1,  m0.y * w);
  atomAddF32(op + 2,  m0.z * w);
  atomAddF32(op + 3,  m0.w * w);
  atomAddF32(op + 4,  m1.x * w);
  atomAddF32(op + 5,  m1.y * w);
  atomAddF32(op + 6,  m1.z * w);
  atomAddF32(op + 7,  m1.w * w);
  atomAddF32(op + 8,  m2.x * w);
  atomAddF32(op + 9,  m2.y * w);
  atomAddF32(op + 10, m2.z * w);
  atomAddF32(op + 11, m2.w * w);
  atomAddF32(op + 12, m3.x * w);
  atomAddF32(op + 13, m3.y * w);
  atomAddF32(op + 14, m3.z * w);
  atomAddF32(op + 15, m3.w * w);
}

// ---- node fuse: h = relu(agg1+b1); h2 = h@W2; out init = h2*dinv^2 + b2 --

__global__ void k_node2(const float* __restrict__ agg1,
                        const float* __restrict__ b1,
                        const float* __restrict__ W2,
                        const float* __restrict__ b2,
                        const float* __restrict__ dinv,
                        float* __restrict__ h2,
                        float* __restrict__ out, int N) {
  int n = blockIdx.x * blockDim.x + threadIdx.x;
  if (n >= N) return;
  const float4* ap = (const float4*)(agg1 + (size_t)n * 16);
  float c0 = 0.0f, c1 = 0.0f;
#pragma unroll
  for (int i = 0; i < 4; ++i) {
    float4 t = ap[i];
    float v[4] = {t.x, t.y, t.z, t.w};
#pragma unroll
    for (int j = 0; j < 4; ++j) {
      const int k = i * 4 + j;
      const float hv = fmaxf(v[j] + b1[k], 0.0f);
      c0 = fmaf(hv, W2[k * 2 + 0], c0);
      c1 = fmaf(hv, W2[k * 2 + 1], c1);
    }
  }
  h2[(size_t)n * 2 + 0] = c0;
  h2[(size_t)n * 2 + 1] = c1;
  const float di = dinv[n];
  const float w = di * di;
  out[(size_t)n * 2 + 0] = fmaf(c0, w, b2[0]);  // self-loop + bias init
  out[(size_t)n * 2 + 1] = fmaf(c1, w, b2[1]);
}

// ---- layer 2 edge aggregation: out[dst] += h2[src] * norm ----------------

__global__ void k_edge_agg2(const int* __restrict__ ei,
                            const float* __restrict__ dinv,
                            const float* __restrict__ h2,
                            float* __restrict__ out, int E) {
  int e = blockIdx.x * blockDim.x + threadIdx.x;
  if (e >= E) return;
  const int s = ei[e];
  const int d = ei[E + e];
  const float w = dinv[s] * dinv[d];
  const float2 m = *(const float2*)(h2 + (size_t)s * 2);
  atomAddF32(out + (size_t)d * 2 + 0, m.x * w);
  atomAddF32(out + (size_t)d * 2 + 1, m.y * w);
}

// ---- launch --------------------------------------------------------------

extern "C" void kernel_launch(void* const* d_in, const int* in_sizes, int n_in,
                              void* d_out, int out_size, void* d_ws, size_t ws_size,
                              hipStream_t stream) {
  const float* x  = (const float*)d_in[0];
  const int*   ei = (const int*)d_in[1];   // [2, E] int32 (JAX default x32)
  const float* W1 = (const float*)d_in[2];
  const float* b1 = (const float*)d_in[3];
  const float* W2 = (const float*)d_in[4];
  const float* b2 = (const float*)d_in[5];
  float* out = (float*)d_out;

  const int N = in_sizes[0] / 128;
  const int E = in_sizes[1] / 2;

  // workspace layout (floats): deg/dinv [N] | h1 [N*16] | agg1 [N*16] | h2 [N*2]
  float* deg  = (float*)d_ws;
  float* h1   = deg + N;
  float* agg1 = h1 + (size_t)N * 16;
  float* h2   = agg1 + (size_t)N * 16;

  dim3 blk(TPB);
  dim3 gN((N + TPB - 1) / TPB);
  dim3 gE((E + TPB - 1) / TPB);

  k_init_deg<<<gN, blk, 0, stream>>>(deg, N);
  k_edge_deg<<<gE, blk, 0, stream>>>(ei + E, deg, E);
  k_dinv<<<gN, blk, 0, stream>>>(deg, N);

  const int tiles = (N + 15) / 16;
  const int wavesPerBlock = TPB / 32;
  dim3 gT((tiles + wavesPerBlock - 1) / wavesPerBlock);
  k_gemm1_wmma<<<gT, blk, 0, stream>>>(x, W1, deg, h1, agg1, N);

  k_edge_agg16<<<gE, blk, 0, stream>>>(ei, deg, h1, agg1, E);
  k_node2<<<gN, blk, 0, stream>>>(agg1, b1, W2, b2, deg, h2, out, N);
  k_edge_agg2<<<gE, blk, 0, stream>>>(ei, deg, h2, out, E);
}